// DescendantMax_3702261809397
// MI455X (gfx1250) — compile-verified
//
#include <hip/hip_runtime.h>
#include <stdint.h>

#define N_NODES 299593
#define BATCH   64

// Balanced 8-ary tree, BFS numbering. Level k starts at s_k=(8^k-1)/7 and has
// 8^k nodes; children of the j-th node of level k are the 8 contiguous nodes
// starting at s_{k+1} + 8*j. So no index arrays are needed.
//
// Level-5 (deepest) sweep, fused with the leaf copy-out:
//  - each block owns 256 parents of level 5 -> their 2048 children (leaves)
//    are one contiguous 8KB span of x.
//  - stage that span into LDS with gfx1250 async global->LDS loads
//    (ASYNCcnt-tracked), aligned down to 16B since row bases are only 4B
//    aligned (N_NODES is odd).
//  - each thread reads its 8 children from LDS, writes the 8 leaf copies to
//    out, and writes out[parent] = max(x[parent], max8(children)).
__global__ __launch_bounds__(256) void desc_max_level5_fused(
    const float* __restrict__ x, float* __restrict__ out) {
  __shared__ __align__(16) float tile[2052];  // 2048 children + <=4 halo

  const int t    = threadIdx.x;
  const int b    = blockIdx.x >> 7;   // 128 blocks per batch row
  const int pblk = blockIdx.x & 127;
  const int P0   = pblk << 8;         // first parent (within level 5) of block

  const size_t   row   = (size_t)b * N_NODES;
  const size_t   C0    = row + 37449u + (size_t)P0 * 8u;  // first child elem
  const unsigned delta = (unsigned)(C0 & 3u);             // misalignment, elems
  const size_t   A0    = C0 - delta;                      // 16B-aligned window
  const unsigned gbyte = (unsigned)(A0 * 4u);             // < 2^31, fits I32

  // 32-bit LDS byte address (low 32 bits of the flat LDS-aperture address).
  const unsigned ldsbase = (unsigned)(uintptr_t)(&tile[0]);
  const unsigned lds_a   = ldsbase + (unsigned)t * 16u;
  const unsigned g_a     = gbyte + (unsigned)t * 16u;

  // 2 x b128 per thread: 256 threads * 32B = 8KB window (INST_OFFSET is added
  // to BOTH the LDS and the global address, so one reg pair serves both).
  asm volatile("global_load_async_to_lds_b128 %0, %1, %2 offset:0"
               :: "v"(lds_a), "v"(g_a), "s"(x) : "memory");
  asm volatile("global_load_async_to_lds_b128 %0, %1, %2 offset:4096"
               :: "v"(lds_a), "v"(g_a), "s"(x) : "memory");
  // Halo: only the 'delta' extra trailing elements are needed; b32 loads keep
  // every global access strictly in-bounds.
  const unsigned lds_r = ldsbase + 8192u + (unsigned)t * 4u;
  const unsigned g_r   = gbyte + 8192u + (unsigned)t * 4u;
  if ((unsigned)t < delta) {
    asm volatile("global_load_async_to_lds_b32 %0, %1, %2 offset:0"
                 :: "v"(lds_r), "v"(g_r), "s"(x) : "memory");
  }
  asm volatile("s_wait_asynccnt 0" ::: "memory");  // own wave's async done
  __syncthreads();                                 // all waves' async done

  const int base = (int)delta + t * 8;
  const float c0 = tile[base + 0], c1 = tile[base + 1];
  const float c2 = tile[base + 2], c3 = tile[base + 3];
  const float c4 = tile[base + 4], c5 = tile[base + 5];
  const float c6 = tile[base + 6], c7 = tile[base + 7];
  const float m = fmaxf(fmaxf(fmaxf(c0, c1), fmaxf(c2, c3)),
                        fmaxf(fmaxf(c4, c5), fmaxf(c6, c7)));

  // Parent of level 5 (each node is written exactly once across the sweep, so
  // its pre-update value is x[parent]).
  const size_t pidx = row + 4681u + (unsigned)(P0 + t);
  out[pidx] = fmaxf(x[pidx], m);

  // Fused leaf copy-out (values already live in registers).
  float* __restrict__ cdst = out + C0 + (size_t)t * 8u;
  cdst[0] = c0; cdst[1] = c1; cdst[2] = c2; cdst[3] = c3;
  cdst[4] = c4; cdst[5] = c5; cdst[6] = c6; cdst[7] = c7;
}

// Levels 4..0: one thread per (batch row, parent). Children were written by
// the previous (stream-ordered) level kernel; parent initial value is
// x[parent]. Tiny traffic (~8.4MB total), scalar loads are fine.
__global__ __launch_bounds__(256) void desc_max_level(
    const float* __restrict__ x, float* __restrict__ out,
    int s_par, int s_child, int n_par_mask, int b_shift) {
  const int tid   = blockIdx.x * blockDim.x + threadIdx.x;
  const int total = BATCH << b_shift;  // BATCH * n_par
  if (tid >= total) return;
  const int j = tid & n_par_mask;
  const int b = tid >> b_shift;
  const size_t row = (size_t)b * N_NODES;

  const float* __restrict__ ch = out + row + (unsigned)s_child + (size_t)j * 8u;
  const float m = fmaxf(fmaxf(fmaxf(ch[0], ch[1]), fmaxf(ch[2], ch[3])),
                        fmaxf(fmaxf(ch[4], ch[5]), fmaxf(ch[6], ch[7])));
  const size_t pidx = row + (unsigned)s_par + (unsigned)j;
  out[pidx] = fmaxf(x[pidx], m);
}

extern "C" void kernel_launch(void* const* d_in, const int* in_sizes, int n_in,
                              void* d_out, int out_size, void* d_ws,
                              size_t ws_size, hipStream_t stream) {
  (void)in_sizes; (void)n_in; (void)out_size; (void)d_ws; (void)ws_size;
  const float* x = (const float*)d_in[0];  // [64, 299593] fp32
  float* out = (float*)d_out;              // [64, 299593] fp32

  // Deepest level (5): fused leaf copy + parent max. 64 rows * 128 blocks.
  desc_max_level5_fused<<<BATCH * 128, 256, 0, stream>>>(x, out);

  // Levels 4..0, bottom-up; stream order provides the dependency chain.
  static const int s[7] = {0, 1, 9, 73, 585, 4681, 37449};
  for (int k = 4; k >= 0; --k) {
    const int shift  = 3 * k;           // n_par = 8^k
    const int total  = BATCH << shift;
    const int blocks = (total + 255) / 256;
    desc_max_level<<<blocks, 256, 0, stream>>>(x, out, s[k], s[k + 1],
                                               (1 << shift) - 1, shift);
  }
}